// ArthDenseCalcToDenseBlock_2319282340020
// MI455X (gfx1250) — compile-verified
//
#include <hip/hip_runtime.h>
#include <math.h>

#define DIV_EPS 1e-7f
#define POW_EPS 1e-7f
#define SMAX 256
#define BLOCK 256

// ---------- CDNA5 async global->LDS path (guarded; fallback = plain loads) ----
#if defined(__HIP_DEVICE_COMPILE__) && __has_builtin(__builtin_amdgcn_global_load_async_to_lds_b32)
#define USE_ASYNC_LDS 1
#else
#define USE_ASYNC_LDS 0
#endif

#if USE_ASYNC_LDS
// Builtin signature (from hipcc diagnostic): param0 is AS1 (global/"__device__")
// int*, param1 is AS3 (LDS) int*, then two immediate ints (offset, cpol).
typedef __attribute__((address_space(1))) int* as1_int_ptr;
typedef __attribute__((address_space(3))) int* as3_int_ptr;

__device__ __forceinline__ void async_g2l_b32(const float* g, float* l) {
  // generic->AS1 is numerically identity; generic->AS3 is the low-32-bit LDS
  // offset (ISA 10.2); integer casts so the
  // address-space conversion always compiles.
  __builtin_amdgcn_global_load_async_to_lds_b32(
      (as1_int_ptr)(unsigned long long)(const void*)g,
      (as3_int_ptr)(unsigned int)(unsigned long long)(const void*)l,
      0, 0);
}

__device__ __forceinline__ void wait_asynccnt0() {
#if __has_builtin(__builtin_amdgcn_s_wait_asynccnt)
  __builtin_amdgcn_s_wait_asynccnt(0);
#else
  asm volatile("s_wait_asynccnt 0" ::: "memory");
#endif
}
#endif

// -----------------------------------------------------------------------------
// One block per batch row b.  Threads: 256 (8 waves, wave32).
//  Phase 0: issue async global->LDS staging of tv/td rows (ASYNCcnt) so the
//           transfers overlap the streaming op-row copy + argmax below.
//  Phase 1: stream-copy op row to out (float4), per-position argmax -> LDS.
//  Phase 2: s_wait_asynccnt 0 + barrier; lane 0 runs the exact sequential
//           interpreter (early-exits once (1-meet)==0: provably frozen state).
//  Phase 3: cooperative store of mutated tv/td rows.
// -----------------------------------------------------------------------------
__global__ __launch_bounds__(BLOCK)
void ArthDenseScan_kernel(const float* __restrict__ g_tv,
                          const float* __restrict__ g_td,
                          const float* __restrict__ g_op,
                          const float* __restrict__ g_fin,
                          const float* __restrict__ g_ifv,
                          const int*   __restrict__ g_start,
                          float* __restrict__ o_tv,
                          float* __restrict__ o_td,
                          float* __restrict__ o_op,
                          float* __restrict__ o_iff,
                          float* __restrict__ o_iv,
                          int S, int NOPS)
{
  __shared__ float s_tv[SMAX];
  __shared__ float s_td[SMAX];
  __shared__ int   s_cls[SMAX];

  const int b = blockIdx.x;
  const int t = threadIdx.x;

  const float* tvrow = g_tv + (size_t)b * S;
  const float* tdrow = g_td + (size_t)b * S;
  const float* oprow = g_op + (size_t)b * S * NOPS;

  const int rowlen = S * NOPS;

  // ---- Phase 0: stage tv/td rows into LDS (CDNA5 async copy, ASYNCcnt) ----
  if (t < S) {
#if USE_ASYNC_LDS
    async_g2l_b32(tvrow + t, &s_tv[t]);
    async_g2l_b32(tdrow + t, &s_td[t]);
#else
    s_tv[t] = tvrow[t];
    s_td[t] = tdrow[t];
#endif
  }

#if defined(__HIP_DEVICE_COMPILE__)
  // gfx1250 global_prefetch_b8 for the streaming op row
  __builtin_prefetch(oprow + (size_t)t * ((rowlen + (BLOCK - 1)) / BLOCK), 0, 0);
#endif

  // ---- Phase 1a: passthrough copy of trans_op row (coalesced, float4) ----
  {
    float* dst = o_op + (size_t)b * rowlen;
    if ((rowlen & 3) == 0) {
      const float4* src4 = reinterpret_cast<const float4*>(oprow);
      float4*       dst4 = reinterpret_cast<float4*>(dst);
      const int n4 = rowlen >> 2;
      for (int i = t; i < n4; i += BLOCK) dst4[i] = src4[i];
    } else {
      for (int i = t; i < rowlen; i += BLOCK) dst[i] = oprow[i];
    }
  }

  // ---- Phase 1b: per-position token class (jnp.argmax: first max wins) ----
  if (t < S) {
    const float* o = oprow + (size_t)t * NOPS;
    float best = o[0];
    int   cls  = 0;
    if (NOPS == 7) {                // actual shape: fully unrolled compare chain
#pragma unroll
      for (int k = 1; k < 7; ++k) {
        float v = o[k];
        if (v > best) { best = v; cls = k; }
      }
    } else {
      for (int k = 1; k < NOPS; ++k) {
        float v = o[k];
        if (v > best) { best = v; cls = k; }
      }
    }
    s_cls[t] = cls;
  }

#if USE_ASYNC_LDS
  wait_asynccnt0();
#endif
  __syncthreads();

  // ---- Phase 2: exact sequential interpreter on the LDS rows ----
  if (t == 0) {
    const float fin = g_fin[b];
    float iv        = g_ifv[b];
    const int sp    = g_start[0];

    float hn0 = 0.f, hn1 = 0.f;   // hold_numbers
    float hs0 = 0.f, hs1 = 0.f;   // hold_status
    int   hi0 = 0,   hi1 = 0;     // hold_index
    float meet = 0.f;

    const bool fin_latch = ((1.f - fin) == 0.f);   // itv == 0 forever

    for (int i = 0; i < S; ++i) {
      if (fin_latch || (1.f - meet) == 0.f) break;  // provably frozen state

      const float active = (i >= sp) ? 1.f : 0.f;
      const float tv_i = s_tv[i];
      const float itv  = tv_i * (1.f - meet) * (1.f - fin) * active;
      if (itv == 0.f) continue;                      // exact no-op step

      const int   c     = s_cls[i];
      const float notop = (c == 0) ? 1.f : 0.f;

      const float dag = iv * notop * itv;
      if (dag != 0.f) {
        // push: h = dag*[0, h0] + (1-dag)*h, then slot0 overwritten
        const float td_i = s_td[i];
        float nhn0 = (1.f - dag) * hn0;
        float nhn1 = dag * hn0 + (1.f - dag) * hn1;
        float nhs0 = (1.f - dag) * hs0;
        float nhs1 = dag * hs0 + (1.f - dag) * hs1;
        const int gi = (int)dag;                     // astype(int32) truncation
        int nhi0 = (1 - gi) * hi0;
        int nhi1 = gi * hi0 + (1 - gi) * hi1;
        nhi0 = nhi0 * (1 - gi) + gi * i;
        nhs0 = nhs0 * (1.f - dag) + dag;
        nhn0 = nhn0 * (1.f - dag) + dag * td_i;
        hn0 = nhn0; hn1 = nhn1; hs0 = nhs0; hs1 = nhs1; hi0 = nhi0; hi1 = nhi1;
      }

      const float op_gate = (1.f - notop) * itv * iv;
      meet = op_gate * (1.f - meet) + meet;
      iv   = op_gate * hs0 * hs1 + (1.f - op_gate) * iv;

      const float fg = (1.f - notop) * itv * iv;     // uses UPDATED iv
      if (fg != 0.f) {
        const float h0 = hn0, h1 = hn1;
        float dense;
        switch (c) {
          case 2:  dense = h1 + h0;                       break;
          case 3:  dense = h1 - h0;                       break;
          case 4:  dense = h1 * h0;                       break;
          case 5:  dense = h1 / (h0 + DIV_EPS);           break;
          case 6:  dense = powf(fmaxf(h1, POW_EPS), h0);  break;
          default: dense = 0.f;                           break;
        }
        s_tv[hi1] *= (1.f - fg);
        s_td[hi0]  = s_td[hi0] * (1.f - fg) + dense * fg;
        s_tv[i]   *= (1.f - fg);
        hs0       *= (1.f - fg);
      }
    }

    const float ifu = iv * (1.f - meet) * hs0 * (1.f - hs1);
    o_iff[b] = (1.f - fin) * ifu + fin;
    o_iv[b]  = iv;
  }
  __syncthreads();

  // ---- Phase 3: store mutated rows ----
  if (t < S) {
    o_tv[(size_t)b * S + t] = s_tv[t];
    o_td[(size_t)b * S + t] = s_td[t];
  }
}

extern "C" void kernel_launch(void* const* d_in, const int* in_sizes, int n_in,
                              void* d_out, int out_size, void* d_ws, size_t ws_size,
                              hipStream_t stream) {
  const float* g_tv    = (const float*)d_in[0];   // trans_valid (B,S)
  const float* g_td    = (const float*)d_in[1];   // trans_dense (B,S)
  const float* g_op    = (const float*)d_in[2];   // trans_op (B,S,NOPS)
  const float* g_fin   = (const float*)d_in[3];   // if_finished (B)
  const float* g_ifv   = (const float*)d_in[4];   // if_valid (B)
  const int*   g_start = (const int*)d_in[5];     // start_pos scalar

  const int B    = in_sizes[3];
  const int S    = in_sizes[0] / B;
  const int NOPS = in_sizes[2] / (B * S);

  float* out   = (float*)d_out;                   // (tv, td, op, iff, iv) concat
  float* o_tv  = out;
  float* o_td  = out + (size_t)B * S;
  float* o_op  = out + (size_t)2 * B * S;
  float* o_iff = out + (size_t)(2 + NOPS) * B * S;
  float* o_iv  = o_iff + B;

  dim3 grid(B), block(BLOCK);
  ArthDenseScan_kernel<<<grid, block, 0, stream>>>(
      g_tv, g_td, g_op, g_fin, g_ifv, g_start,
      o_tv, o_td, o_op, o_iff, o_iv, S, NOPS);
}